// Cell_6631429505481
// MI455X (gfx1250) — compile-verified
//
#include <hip/hip_runtime.h>
#include <math.h>

// ---------------------------------------------------------------------------
// CDNA5 (gfx1250) implementation of the GNN NAS-cell reference.
// GEMMs run on V_WMMA_F32_16X16X4_F32 (fp32 matrix pipe, wave32).
// ---------------------------------------------------------------------------

typedef __attribute__((ext_vector_type(2))) float v2f;
typedef __attribute__((ext_vector_type(8))) float v8f;

#define KCAND 3
#define EPS_N 1e-5f

// Monotonic uint encoding of float for atomicMax-based segment max.
__device__ __forceinline__ unsigned enc_f32(float f) {
  unsigned u = __float_as_uint(f);
  return (u & 0x80000000u) ? ~u : (u | 0x80000000u);
}
__device__ __forceinline__ float dec_f32(unsigned k) {
  return (k & 0x80000000u) ? __uint_as_float(k & 0x7fffffffu)
                           : __uint_as_float(~k);
}

// ---- utility: zero a float range ------------------------------------------
__global__ void k_zero(float* p, int n) {
  int i = blockIdx.x * blockDim.x + threadIdx.x;
  if (i < n) p[i] = 0.f;
}

// ---- graph aggregation (segment mean + segment max) ------------------------
__global__ void k_agg_init(float* sum, unsigned* maxk, float* deg, int nND, int nN) {
  int i = blockIdx.x * blockDim.x + threadIdx.x;
  if (i < nND) { sum[i] = 0.f; maxk[i] = 0x007FFFFFu; /* enc(-inf) */ }
  if (i < nN) deg[i] = 0.f;
}

__global__ void k_agg_deg(const int* __restrict__ dst, float* deg, int E) {
  int e = blockIdx.x * blockDim.x + threadIdx.x;
  if (e < E) atomicAdd(&deg[dst[e]], 1.f);
}

// one edge handled by 32 threads, 4 features each (float4 loads)
__global__ void k_agg_scatter(const float* __restrict__ h, const int* __restrict__ src,
                              const int* __restrict__ dst, float* sum, unsigned* maxk,
                              int E) {
  int t = blockIdx.x * blockDim.x + threadIdx.x;
  int e = t >> 5;
  if (e >= E) return;
  int d = (t & 31) * 4;
  int s = src[e], q = dst[e];
  const float4 v = *(const float4*)(h + (size_t)s * 128 + d);
  float* sp = sum + (size_t)q * 128 + d;
  unsigned* mp = maxk + (size_t)q * 128 + d;
  atomicAdd(sp + 0, v.x); atomicAdd(sp + 1, v.y);
  atomicAdd(sp + 2, v.z); atomicAdd(sp + 3, v.w);
  atomicMax(mp + 0, enc_f32(v.x)); atomicMax(mp + 1, enc_f32(v.y));
  atomicMax(mp + 2, enc_f32(v.z)); atomicMax(mp + 3, enc_f32(v.w));
}

__global__ void k_agg_finalize(float* sum, unsigned* maxk, const float* __restrict__ deg,
                               int nND) {
  int i = blockIdx.x * blockDim.x + threadIdx.x;
  if (i >= nND) return;
  float dg = fmaxf(deg[i >> 7], 1.f);
  sum[i] = sum[i] / dg;
  float m = dec_f32(maxk[i]);
  ((float*)maxk)[i] = isfinite(m) ? m : 0.f;   // empty segments -> 0
}

// ---- mixed op pass 1: 3 GEMMs via fp32 WMMA + per-(k,d) sum/sumsq ----------
// Block = 256 threads (8 waves). Block covers 16 rows; wave w covers columns
// [16w,16w+16). preMode: candidates computed from (src_emb, hr) on the fly.
__global__ void k_mixed_gemm(const float* __restrict__ p0, const float* __restrict__ p1,
                             const float* __restrict__ p2, int preMode,
                             const float* __restrict__ W, const float* __restrict__ b,
                             float* __restrict__ y, float* sum, float* sumsq, int nRows) {
  __shared__ float As[16][132];       // +4 pad: conflict-free column reads
  const int tid = threadIdx.x;
  const int wave = tid >> 5;
  const int lane = tid & 31;
  const int m = lane & 15;            // row of A / col of B held by this lane
  const int hh = lane >> 4;           // half-wave selector
  const int rowbase = blockIdx.x * 16;
  const int colbase = wave * 16;

  for (int k = 0; k < KCAND; ++k) {
    __syncthreads();
    // stage candidate-k tile (16 x 128) into LDS
    for (int idx = tid; idx < 16 * 128; idx += 256) {
      int r = idx >> 7, c = idx & 127;
      int row = rowbase + r;
      float v = 0.f;
      if (row < nRows) {
        if (preMode) {
          float s = p0[(size_t)row * 128 + c];
          float t = p1[(size_t)row * 128 + c];
          v = (k == 0) ? s : ((k == 1) ? (s - t) : (s * t));
        } else {
          const float* xk = (k == 0) ? p0 : ((k == 1) ? p1 : p2);
          v = xk[(size_t)row * 128 + c];
        }
      }
      As[r][c] = v;
    }
    __syncthreads();

    const float* Wk = W + (size_t)k * 128 * 128;
    v8f acc = {};
    // 16x16 tile = 32 x V_WMMA_F32_16X16X4_F32 along K=128
    for (int kk = 0; kk < 128; kk += 4) {
      v2f av, bv;
      av.x = As[m][kk + 2 * hh];                            // A[m][2h], A[m][2h+1]
      av.y = As[m][kk + 2 * hh + 1];
      bv.x = Wk[(size_t)(kk + 2 * hh) * 128 + colbase + m]; // B[2h][n], B[2h+1][n]
      bv.y = Wk[(size_t)(kk + 2 * hh + 1) * 128 + colbase + m];
      acc = __builtin_amdgcn_wmma_f32_16x16x4_f32(false, av, false, bv,
                                                  (short)0, acc, false, false);
    }

    // bias, store y, accumulate batch-norm statistics over N
    float bias = b[k * 128 + colbase + m];
    float ls = 0.f, lq = 0.f;
#pragma unroll
    for (int r = 0; r < 8; ++r) {
      int row = rowbase + r + 8 * hh;   // C/D layout: row = r + 8h, col = m
      float v = acc[r] + bias;
      if (row < nRows) {
        y[(size_t)k * nRows * 128 + (size_t)row * 128 + colbase + m] = v;
        ls += v; lq += v * v;
      }
    }
    atomicAdd(&sum[k * 128 + colbase + m], ls);
    atomicAdd(&sumsq[k * 128 + colbase + m], lq);
  }
}

// ---- mixed op pass 2: normalize over N, relu, weighted sum over K ----------
__global__ void k_mixed_norm(const float* __restrict__ y, const float* __restrict__ sum,
                             const float* __restrict__ sumsq, const float* __restrict__ g,
                             const float* __restrict__ be, const float* __restrict__ w,
                             float* out, int accum, int nRows) {
  int i = blockIdx.x * blockDim.x + threadIdx.x;
  if (i >= nRows * 128) return;
  int d = i & 127;
  float invN = 1.f / (float)nRows;
  size_t sz = (size_t)nRows * 128;
  float a = accum ? out[i] : 0.f;
#pragma unroll
  for (int k = 0; k < KCAND; ++k) {
    float mu = sum[k * 128 + d] * invN;
    float var = sumsq[k * 128 + d] * invN - mu * mu;
    float t = (y[(size_t)k * sz + i] - mu) * rsqrtf(var + EPS_N) * g[k * 128 + d]
              + be[k * 128 + d];
    a += w[k] * fmaxf(t, 0.f);
  }
  out[i] = a;
}

// ---- concat GEMM: [m0|m1|s_last](Nx384) @ W_cat(384x128) + bias + stats ----
__global__ void k_cat_gemm(const float* __restrict__ a0, const float* __restrict__ a1,
                           const float* __restrict__ a2, const float* __restrict__ Wc,
                           const float* __restrict__ bc, float* __restrict__ y,
                           float* sum, float* sumsq, int nRows) {
  __shared__ float As[16][388];       // 16 x 384 (+pad)
  const int tid = threadIdx.x;
  const int wave = tid >> 5;
  const int lane = tid & 31;
  const int m = lane & 15;
  const int hh = lane >> 4;
  const int rowbase = blockIdx.x * 16;
  const int colbase = wave * 16;

  for (int idx = tid; idx < 16 * 384; idx += 256) {
    int r = idx / 384, c = idx - r * 384;
    int row = rowbase + r;
    float v = 0.f;
    if (row < nRows) {
      const float* src = (c < 128) ? a0 : ((c < 256) ? a1 : a2);
      v = src[(size_t)row * 128 + (c & 127)];
    }
    As[r][c] = v;
  }
  __syncthreads();

  v8f acc = {};
  for (int kk = 0; kk < 384; kk += 4) {
    v2f av, bv;
    av.x = As[m][kk + 2 * hh];
    av.y = As[m][kk + 2 * hh + 1];
    bv.x = Wc[(size_t)(kk + 2 * hh) * 128 + colbase + m];
    bv.y = Wc[(size_t)(kk + 2 * hh + 1) * 128 + colbase + m];
    acc = __builtin_amdgcn_wmma_f32_16x16x4_f32(false, av, false, bv,
                                                (short)0, acc, false, false);
  }

  float bias = bc[colbase + m];
  float ls = 0.f, lq = 0.f;
#pragma unroll
  for (int r = 0; r < 8; ++r) {
    int row = rowbase + r + 8 * hh;
    float v = acc[r] + bias;
    if (row < nRows) {
      y[(size_t)row * 128 + colbase + m] = v;
      ls += v; lq += v * v;
    }
  }
  atomicAdd(&sum[colbase + m], ls);
  atomicAdd(&sumsq[colbase + m], lq);
}

__global__ void k_cat_norm(const float* __restrict__ y, const float* __restrict__ sum,
                           const float* __restrict__ sumsq, const float* __restrict__ g,
                           const float* __restrict__ be, float* out, int nRows) {
  int i = blockIdx.x * blockDim.x + threadIdx.x;
  if (i >= nRows * 128) return;
  int d = i & 127;
  float invN = 1.f / (float)nRows;
  float mu = sum[d] * invN;
  float var = sumsq[d] * invN - mu * mu;
  float t = (y[i] - mu) * rsqrtf(var + EPS_N) * g[d] + be[d];
  out[i] = fmaxf(t, 0.f);
}

// ---------------------------------------------------------------------------
extern "C" void kernel_launch(void* const* d_in, const int* in_sizes, int n_in,
                              void* d_out, int out_size, void* d_ws, size_t ws_size,
                              hipStream_t stream) {
  const int N = 100000, E = 600000;
  const size_t ND = (size_t)N * 128;
  const size_t WSTRIDE = (size_t)3 * 128 * 128;  // per-op weight stack stride
  const size_t BSTRIDE = 3 * 128;                // per-op bias/gamma/beta stride

  const float* src_emb = (const float*)d_in[0];
  const float* hr      = (const float*)d_in[1];
  const int*   e_src   = (const int*)d_in[2];
  const int*   e_dst   = (const int*)d_in[3];
  const float* w_zero  = (const float*)d_in[4];
  const float* w_first = (const float*)d_in[5];
  const float* w_mid   = (const float*)d_in[6];
  const float* w_last  = (const float*)d_in[7];
  const float* W_zero  = (const float*)d_in[8];
  const float* b_zero  = (const float*)d_in[9];
  const float* g_zero  = (const float*)d_in[10];
  const float* be_zero = (const float*)d_in[11];
  const float* W_first = (const float*)d_in[12];
  const float* b_first = (const float*)d_in[13];
  const float* g_first = (const float*)d_in[14];
  const float* be_first= (const float*)d_in[15];
  const float* W_mid   = (const float*)d_in[16];
  const float* b_mid   = (const float*)d_in[17];
  const float* g_mid   = (const float*)d_in[18];
  const float* be_mid  = (const float*)d_in[19];
  const float* W_last  = (const float*)d_in[20];
  const float* b_last  = (const float*)d_in[21];
  const float* g_last  = (const float*)d_in[22];
  const float* be_last = (const float*)d_in[23];
  const float* W_cat   = (const float*)d_in[24];
  const float* b_cat   = (const float*)d_in[25];
  const float* g_cat   = (const float*)d_in[26];
  const float* be_cat  = (const float*)d_in[27];
  float* out = (float*)d_out;

  // workspace carve-up (floats)
  float* wsf = (float*)d_ws;
  float* y3      = wsf;                 // 3*ND  : mixed-op y scratch
  float* B0      = y3 + 3 * ND;         // ND    : h_in -> m0
  float* B1      = B0 + ND;             // ND    : s0   -> m1
  float* B2      = B1 + ND;             // ND    : s1   -> s_last
  float* ag1_sum = B2 + ND;             // ND
  unsigned* ag1_max = (unsigned*)(ag1_sum + ND);   // ND
  float* ag1_deg = (float*)ag1_max + ND;           // N
  float* ag2_sum = ag1_deg + N;                    // ND
  unsigned* ag2_max = (unsigned*)(ag2_sum + ND);   // ND
  float* ag2_deg = (float*)ag2_max + ND;           // N
  float* stats   = ag2_deg + N;         // 384 (sum)
  float* statsq  = stats + 384;         // 384 (sumsq)

  dim3 blk(256);
  const int gND   = (int)((ND + 255) / 256);     // 50000
  const int gRows = N / 16;                       // 6250 (N % 16 == 0)
  const int gE    = (E + 255) / 256;
  const int gE32  = (int)(((size_t)E * 32 + 255) / 256);

  auto run_mixed = [&](const float* c0, const float* c1, const float* c2, int pre,
                       const float* W, const float* b, const float* g, const float* be,
                       const float* w, float* o, int accum) {
    k_zero<<<4, blk, 0, stream>>>(stats, 768);
    k_mixed_gemm<<<gRows, blk, 0, stream>>>(c0, c1, c2, pre, W, b, y3, stats, statsq, N);
    k_mixed_norm<<<gND, blk, 0, stream>>>(y3, stats, statsq, g, be, w, o, accum, N);
  };
  auto run_agg = [&](const float* h, float* asum, unsigned* amax, float* adeg) {
    k_agg_init<<<gND, blk, 0, stream>>>(asum, amax, adeg, (int)ND, N);
    k_agg_deg<<<gE, blk, 0, stream>>>(e_dst, adeg, E);
    k_agg_scatter<<<gE32, blk, 0, stream>>>(h, e_src, e_dst, asum, amax, E);
    k_agg_finalize<<<gND, blk, 0, stream>>>(asum, amax, adeg, (int)ND);
  };

  // h_in = mixed(pre(src_emb,hr), zero)
  run_mixed(src_emb, hr, nullptr, 1, W_zero, b_zero, g_zero, be_zero, w_zero, B0, 0);
  // agg(h_in) -> set1
  run_agg(B0, ag1_sum, ag1_max, ag1_deg);
  // s0 = mixed(cands(h_in), first[0])
  run_mixed(B0, ag1_sum, (float*)ag1_max, 0, W_first, b_first, g_first, be_first,
            w_first, B1, 0);
  // agg(s0) -> set2
  run_agg(B1, ag2_sum, ag2_max, ag2_deg);
  // s1 = mixed(cands(h_in), first[1]) + mixed(cands(s0), first[2])
  run_mixed(B0, ag1_sum, (float*)ag1_max, 0, W_first + WSTRIDE, b_first + BSTRIDE,
            g_first + BSTRIDE, be_first + BSTRIDE, w_first + 3, B2, 0);
  run_mixed(B1, ag2_sum, (float*)ag2_max, 0, W_first + 2 * WSTRIDE, b_first + 2 * BSTRIDE,
            g_first + 2 * BSTRIDE, be_first + 2 * BSTRIDE, w_first + 6, B2, 1);
  // agg(s1) -> set1 (agg(h_in) dead)
  run_agg(B2, ag1_sum, ag1_max, ag1_deg);
  // m0 = mixed(cands(s0), mid[0]) -> B0 (h_in dead)
  run_mixed(B1, ag2_sum, (float*)ag2_max, 0, W_mid, b_mid, g_mid, be_mid, w_mid, B0, 0);
  // m1 = mixed(cands(s1), mid[1]) -> B1 (s0 dead)
  run_mixed(B2, ag1_sum, (float*)ag1_max, 0, W_mid + WSTRIDE, b_mid + BSTRIDE,
            g_mid + BSTRIDE, be_mid + BSTRIDE, w_mid + 3, B1, 0);
  // agg(m0) -> set2, agg(m1) -> set1
  run_agg(B0, ag2_sum, ag2_max, ag2_deg);
  run_agg(B1, ag1_sum, ag1_max, ag1_deg);
  // s_last = mixed(cands(m0), last[0]) + mixed(cands(m1), last[1]) -> B2 (s1 dead)
  run_mixed(B0, ag2_sum, (float*)ag2_max, 0, W_last, b_last, g_last, be_last,
            w_last, B2, 0);
  run_mixed(B1, ag1_sum, (float*)ag1_max, 0, W_last + WSTRIDE, b_last + BSTRIDE,
            g_last + BSTRIDE, be_last + BSTRIDE, w_last + 3, B2, 1);
  // h = relu(BN_N([m0|m1|s_last] @ W_cat + b_cat))
  k_zero<<<4, blk, 0, stream>>>(stats, 768);
  k_cat_gemm<<<gRows, blk, 0, stream>>>(B0, B1, B2, W_cat, b_cat, y3, stats, statsq, N);
  k_cat_norm<<<gND, blk, 0, stream>>>(y3, stats, statsq, g_cat, be_cat, out, N);
}